// LightGCN_70308614635753
// MI455X (gfx1250) — compile-verified
//
#include <hip/hip_runtime.h>
#include <hip/hip_bf16.h>

// ---------------- problem constants (from reference) ----------------
#define NU   100000              // users
#define NI   50000               // items
#define NN   150000              // nodes
#define EMB  64
#define NE   4800000             // edges
#define KH   3                   // hops

// d_out layout (float4 units): [acc_u | pass_u | acc_i | pass_i]
#define OUT_PASSU4 (NU * 16)              // 1,600,000
#define OUT_ACCI4  (2 * NU * 16)          // 3,200,000
#define OUT_PASSI4 (2 * NU * 16 + NI*16)  // 4,000,000

typedef __attribute__((ext_vector_type(2))) float v2f;
typedef __attribute__((ext_vector_type(8))) float v8f;

// ---- CDNA5 async global->LDS path (guarded; falls back to normal loads) ----
#if defined(__AMDGCN__) &&                                              \
    __has_builtin(__builtin_amdgcn_global_load_async_to_lds_b128) &&    \
    __has_builtin(__builtin_amdgcn_global_load_async_to_lds_b64) &&     \
    __has_builtin(__builtin_amdgcn_s_wait_asynccnt)
#define USE_ASYNC_LDS 1
#else
#define USE_ASYNC_LDS 0
#endif

#if USE_ASYNC_LDS
typedef int v4i_vs __attribute__((vector_size(16)));   // b128 payload type
typedef int v2i_vs __attribute__((vector_size(8)));    // b64 payload type
#define AS1P(T) __attribute__((address_space(1))) T*
#define AS3P(T) __attribute__((address_space(3))) T*

__device__ __forceinline__ void async_b128(const void* gsrc, void* ldst) {
  __builtin_amdgcn_global_load_async_to_lds_b128(
      (AS1P(v4i_vs))gsrc, (AS3P(v4i_vs))ldst, 0, 0);
}
__device__ __forceinline__ void async_b64(const void* gsrc, void* ldst) {
  __builtin_amdgcn_global_load_async_to_lds_b64(
      (AS1P(v2i_vs))gsrc, (AS3P(v2i_vs))ldst, 0, 0);
}
#endif

// ---------------- init: curr = emb0, acc = emb0, passthrough outputs ----
__global__ __launch_bounds__(256)
void lgcn_init(const float4* __restrict__ users4, const float4* __restrict__ items4,
               float4* __restrict__ curr4, float4* __restrict__ out4) {
  int idx = blockIdx.x * blockDim.x + threadIdx.x;   // float4 index over nodes
  if (idx >= NN * 16) return;
  if (idx < NU * 16) {
    float4 v = users4[idx];
    curr4[idx] = v;
    out4[idx] = v;                     // acc_u seed
    out4[OUT_PASSU4 + idx] = v;        // users_emb passthrough
  } else {
    int j = idx - NU * 16;
    float4 v = items4[j];
    curr4[idx] = v;
    out4[OUT_ACCI4 + j] = v;           // acc_i seed
    out4[OUT_PASSI4 + j] = v;          // items_emb passthrough
  }
}

// ---------------- zero the "next" buffer --------------------------------
__global__ __launch_bounds__(256)
void lgcn_zero(float4* __restrict__ p) {
  int idx = blockIdx.x * blockDim.x + threadIdx.x;
  if (idx < NN * 16) p[idx] = make_float4(0.f, 0.f, 0.f, 0.f);
}

// ---------------- SpMM ---------------------------------------------------
// One 256-thread block per 512-edge tile.  Edge metadata (rows/cols/weights,
// 10 KB) is staged into LDS with async global->LDS copies (3 per thread,
// uniform), then 16-lane groups stream 32 edges each: broadcast ds_load of
// metadata, float4 gather from x (L2-resident, 38.4 MB), 4 agent-scope f32
// atomic adds into y (also L2-resident).
#define TILE 512                         // edges per block; NE % TILE == 0

__global__ __launch_bounds__(256)
void lgcn_spmm(const long long* __restrict__ rows, const long long* __restrict__ cols,
               const float* __restrict__ wgt, const float* __restrict__ x,
               float* __restrict__ y) {
  __shared__ long long s_rows[TILE];     // 4096 B
  __shared__ long long s_cols[TILE];     // 4096 B
  __shared__ float     s_w[TILE];        // 2048 B

  const int  tid  = threadIdx.x;
  const long base = (long)blockIdx.x * TILE;

#if USE_ASYNC_LDS
  // 256 threads x (2 x b128 + 1 x b64) covers 4096+4096+2048 bytes exactly.
  async_b128((const char*)(rows + base) + tid * 16, (char*)s_rows + tid * 16);
  async_b128((const char*)(cols + base) + tid * 16, (char*)s_cols + tid * 16);
  async_b64 ((const char*)(wgt  + base) + tid * 8,  (char*)s_w    + tid * 8);
  __builtin_amdgcn_s_wait_asynccnt(0);   // my copies landed in LDS
  __syncthreads();                       // everyone's copies landed
#else
  for (int i = tid; i < TILE; i += 256) {
    s_rows[i] = rows[base + i];
    s_cols[i] = cols[base + i];
    s_w[i]    = wgt[base + i];
  }
  __syncthreads();
#endif

  const int lane = tid & 15;
  const int g    = tid >> 4;             // 16 groups of 16 lanes
  for (int j = 0; j < TILE / 16; ++j) {  // 32 edges per group
    const int   le = g * (TILE / 16) + j;
    const int   r  = (int)s_rows[le];    // broadcast LDS reads
    const int   c  = (int)s_cols[le];
    const float w  = s_w[le];
    const float4 xv = ((const float4*)(x + (size_t)c * EMB))[lane];
    float* dst = y + (size_t)r * EMB + lane * 4;
    (void)__hip_atomic_fetch_add(dst + 0, w * xv.x, __ATOMIC_RELAXED, __HIP_MEMORY_SCOPE_AGENT);
    (void)__hip_atomic_fetch_add(dst + 1, w * xv.y, __ATOMIC_RELAXED, __HIP_MEMORY_SCOPE_AGENT);
    (void)__hip_atomic_fetch_add(dst + 2, w * xv.z, __ATOMIC_RELAXED, __HIP_MEMORY_SCOPE_AGENT);
    (void)__hip_atomic_fetch_add(dst + 3, w * xv.w, __ATOMIC_RELAXED, __HIP_MEMORY_SCOPE_AGENT);
  }
}

// ---------------- acc += next (acc split across d_out slots) ------------
__global__ __launch_bounds__(256)
void lgcn_accadd(const float4* __restrict__ nxt4, float4* __restrict__ out4) {
  int idx = blockIdx.x * blockDim.x + threadIdx.x;
  if (idx >= NN * 16) return;
  float4 v = nxt4[idx];
  float4* dst = (idx < NU * 16) ? (out4 + idx)
                                : (out4 + OUT_ACCI4 + (idx - NU * 16));
  float4 o = *dst;
  o.x += v.x; o.y += v.y; o.z += v.z; o.w += v.w;
  *dst = o;
}

// ---------------- finalize: acc *= 0.25 via chained f32 WMMA ------------
// One wave32 per 16x16 tile.  D = X * (0.25 * I16), split into four K=4
// chunks: D = sum_k A_k(16x4) * B_k(4x16), with A_k = X[:,4k..4k+3] in the
// documented 16x4 f32 A-layout, B_k = 0.25 * rows 4k..4k+3 of I16.
#define ROWTILES_U (NU / 16)        // 6250
#define ROWTILES   (NN / 16)        // 9375
#define NTILES     (ROWTILES * 4)   // 37500

__global__ __launch_bounds__(256)
void lgcn_finalize_wmma(float* __restrict__ out) {
  int l = threadIdx.x & 31;
  int t = (blockIdx.x * blockDim.x + threadIdx.x) >> 5;   // global wave id
  if (t >= NTILES) return;                                 // wave-uniform branch
  int rt = t >> 2, ct = t & 3;
  float* base;
  if (rt < ROWTILES_U) {
    base = out + (size_t)rt * 16 * EMB + (size_t)ct * 16;                       // acc_u
  } else {
    base = out + (size_t)2 * NU * EMB
               + (size_t)(rt - ROWTILES_U) * 16 * EMB + (size_t)ct * 16;        // acc_i
  }

  const int row  = l & 15;           // A: M = lane & 15
  const int kb   = (l >> 4) << 1;    // A: lanes 0-15 hold K={0,1}, 16-31 hold K={2,3}
  const int bcol = l & 15;           // B: N = lane & 15

  v8f c = {};
#pragma unroll
  for (int k = 0; k < 4; ++k) {
    v2f a;
    a.x = base[(size_t)row * EMB + 4 * k + kb + 0];
    a.y = base[(size_t)row * EMB + 4 * k + kb + 1];
    // B_k[r][n] = 0.25 * (n == 4k + r); VGPR v holds rows v (lanes 0-15) and v+2 (16-31)
    v2f b;
    b.x = (bcol == (4 * k + kb + 0)) ? 0.25f : 0.0f;
    b.y = (bcol == (4 * k + kb + 1)) ? 0.25f : 0.0f;
    c = __builtin_amdgcn_wmma_f32_16x16x4_f32(false, a, false, b, (short)0, c,
                                              false, false);
  }

  const int rbase = (l >> 4) << 3;   // D: VGPR j holds M=j (lanes 0-15) / M=j+8 (16-31)
#pragma unroll
  for (int j = 0; j < 8; ++j)
    base[(size_t)(rbase + j) * EMB + (l & 15)] = c[j];
}

// ---------------- host side --------------------------------------------
extern "C" void kernel_launch(void* const* d_in, const int* in_sizes, int n_in,
                              void* d_out, int out_size, void* d_ws, size_t ws_size,
                              hipStream_t stream) {
  const float4*    users4 = (const float4*)d_in[0];
  const float4*    items4 = (const float4*)d_in[1];
  const long long* eidx   = (const long long*)d_in[2];   // int64 (2, NE)
  const float*     ew     = (const float*)d_in[3];
  float* out = (float*)d_out;

  float* bufA = (float*)d_ws;                      // 38.4 MB
  float* bufB = bufA + (size_t)NN * EMB;           // 38.4 MB  (ws total 76.8 MB)

  const long long* rows = eidx;                    // edge_index[0]
  const long long* cols = eidx + NE;               // edge_index[1]

  const int n4       = NN * 16;                    // 2.4M float4
  const int blocksN4 = (n4 + 255) / 256;           // 9375

  lgcn_init<<<blocksN4, 256, 0, stream>>>(users4, items4, (float4*)bufA, (float4*)out);

  float* curr = bufA;
  float* nxt  = bufB;
  const int spmmBlocks = NE / TILE;                // 9375 tiles of 512 edges
  for (int h = 0; h < KH; ++h) {
    lgcn_zero<<<blocksN4, 256, 0, stream>>>((float4*)nxt);
    lgcn_spmm<<<spmmBlocks, 256, 0, stream>>>(rows, cols, ew, curr, nxt);
    lgcn_accadd<<<blocksN4, 256, 0, stream>>>((const float4*)nxt, (float4*)out);
    float* tmp = curr; curr = nxt; nxt = tmp;      // ping-pong
  }

  lgcn_finalize_wmma<<<(NTILES + 7) / 8, 256, 0, stream>>>(out);
}